// EncoderDecoderLSTM_58222576664600
// MI455X (gfx1250) — compile-verified
//
#include <hip/hip_runtime.h>
#include <hip/hip_bf16.h>

typedef float v2f __attribute__((ext_vector_type(2)));
typedef float v8f __attribute__((ext_vector_type(8)));

#define HID   100
#define G4    400           // 4*HID
#define T_ENC 4096
#define T_DEC 4096
#define DEC_STEPS 4095      // decoder produces logits for rows 0..4094
#define NCLS  6

// ---------------------------------------------------------------------------
// Kernel 1: gx_enc[t, j] = x[t,:3] . enc_Wih[j,:3] + enc_bih[j]
// WMMA f32 16x16x4 GEMM, M=4096 (time), N=400 (gates), K=3 padded to 4.
// One wave per 16x16 output tile; 256 * 25 = 6400 waves = 800 blocks of 256.
// ---------------------------------------------------------------------------
__global__ __launch_bounds__(256)
void k_gx_enc(const float* __restrict__ x, const float* __restrict__ Wih,
              const float* __restrict__ bih, float* __restrict__ gx)
{
  int wave = (blockIdx.x * blockDim.x + threadIdx.x) >> 5;
  int lane = threadIdx.x & 31;
  int mt = wave / 25;                 // 0..255 m-tiles
  int nt = wave % 25;                 // 0..24  n-tiles
  if (mt >= T_ENC / 16) return;       // wave-uniform; never taken at this grid
  int half = lane >> 4;               // A/B halves: K={0,1} vs K={2,3}
  int l = lane & 15;
  int m0 = mt * 16, n0 = nt * 16;

  v2f A, B;
  if (!half) { A.x = x[(m0 + l) * 3 + 0]; A.y = x[(m0 + l) * 3 + 1]; }
  else       { A.x = x[(m0 + l) * 3 + 2]; A.y = 0.f; }              // K=3 pad
  if (!half) { B.x = Wih[(n0 + l) * 3 + 0]; B.y = Wih[(n0 + l) * 3 + 1]; }
  else       { B.x = Wih[(n0 + l) * 3 + 2]; B.y = 0.f; }

  v8f c = {};
  v8f d = __builtin_amdgcn_wmma_f32_16x16x4_f32(false, A, false, B,
                                                (short)0, c, false, false);
  float bv = bih[n0 + l];
  #pragma unroll
  for (int r = 0; r < 8; ++r) {
    int m = m0 + r + half * 8;
    gx[m * G4 + n0 + l] = d[r] + bv;
  }
}

// ---------------------------------------------------------------------------
// Kernel 2: gx_dec[t, j] = float(y[t]) * dec_Wih[j] + dec_bih[j]
// (teacher_forcing_ratio == 1  =>  decoder inputs are y[0..4094])
// ---------------------------------------------------------------------------
__global__ __launch_bounds__(256)
void k_gx_dec(const int* __restrict__ y, const float* __restrict__ Wih,
              const float* __restrict__ bih, float* __restrict__ gx)
{
  int idx = blockIdx.x * blockDim.x + threadIdx.x;
  if (idx >= DEC_STEPS * G4) return;
  int t = idx / G4, j = idx - t * G4;
  gx[idx] = (float)y[t] * Wih[j] + bih[j];
}

// ---------------------------------------------------------------------------
// Kernel 3: persistent single-workgroup LSTM recurrence (enc 4096 + dec 4095).
// 512 threads = 16 wave32 on one WGP. Thread t (<400) owns Whh row t in VGPRs
// and bhh[t]; h[100] + activated gates[400] in LDS; c[j] in thread j's regs.
// Decoder h states are dumped to Hdec for the off-critical-path logits GEMM.
// ---------------------------------------------------------------------------
__device__ __forceinline__ float fast_sigmoid(float v) {
  return 1.f / (1.f + __expf(-v));
}
__device__ __forceinline__ float fast_tanh(float v) {
  float a = fabsf(v);
  float e = __expf(-2.f * a);
  float t = (1.f - e) / (1.f + e);
  return copysignf(t, v);
}

__global__ __launch_bounds__(512, 1)
void k_lstm_seq(const float* __restrict__ gxEnc, const float* __restrict__ gxDec,
                const float* __restrict__ encWhh, const float* __restrict__ encBhh,
                const float* __restrict__ decWhh, const float* __restrict__ decBhh,
                float* __restrict__ Hdec)
{
  __shared__ float h_s[HID];
  __shared__ float g_s[G4];
  const int t = threadIdx.x;

  float w[HID];
  float bhh = 0.f;
  float c = 0.f;                       // valid for t < HID

  if (t < HID) h_s[t] = 0.f;
  if (t < G4) {
    #pragma unroll
    for (int k = 0; k < HID; ++k) w[k] = encWhh[t * HID + k];
    bhh = encBhh[t];
  }
  __syncthreads();

  for (int phase = 0; phase < 2; ++phase) {
    const float* gx = (phase == 0) ? gxEnc : gxDec;
    const int nsteps = (phase == 0) ? T_ENC : DEC_STEPS;

    for (int step = 0; step < nsteps; ++step) {
      if (t < G4) {
        float gv = gx[step * G4 + t];          // L2-resident stream, issued early
        float a0 = 0.f, a1 = 0.f, a2 = 0.f, a3 = 0.f;
        #pragma unroll
        for (int k = 0; k < HID; k += 4) {     // 100 FMAs, h broadcast from LDS
          a0 = fmaf(w[k + 0], h_s[k + 0], a0);
          a1 = fmaf(w[k + 1], h_s[k + 1], a1);
          a2 = fmaf(w[k + 2], h_s[k + 2], a2);
          a3 = fmaf(w[k + 3], h_s[k + 3], a3);
        }
        float acc = ((a0 + a1) + (a2 + a3)) + gv + bhh;
        float act = (t >= 2 * HID && t < 3 * HID) ? fast_tanh(acc)      // g
                                                  : fast_sigmoid(acc);  // i,f,o
        g_s[t] = act;
      }
      __syncthreads();
      if (t < HID) {
        float iv = g_s[t], fv = g_s[HID + t], gv = g_s[2 * HID + t],
              ov = g_s[3 * HID + t];
        c = fmaf(fv, c, iv * gv);
        float hn = ov * fast_tanh(c);
        h_s[t] = hn;
        if (phase == 1) Hdec[step * HID + t] = hn;
      }
      __syncthreads();
    }

    if (phase == 0) {                  // swap to decoder weights; h,c carry over
      if (t < G4) {
        #pragma unroll
        for (int k = 0; k < HID; ++k) w[k] = decWhh[t * HID + k];
        bhh = decBhh[t];
      }
    }
  }
}

// ---------------------------------------------------------------------------
// Kernel 4: out[m, n] = Hdec[m,:] . lin_W[n,:] + lin_b[n]  (n < 6), row 4095 = 0
// WMMA f32 16x16x4, K=100 = 25 chained WMMAs per 16x16 tile; N padded 6->16.
// 256 m-tiles, 1 wave each -> 32 blocks of 256.
// ---------------------------------------------------------------------------
__global__ __launch_bounds__(256)
void k_logits(const float* __restrict__ Hd, const float* __restrict__ linW,
              const float* __restrict__ linB, float* __restrict__ out)
{
  int wave = (blockIdx.x * blockDim.x + threadIdx.x) >> 5;  // m-tile 0..255
  int lane = threadIdx.x & 31;
  int half = lane >> 4, l = lane & 15;
  int m0 = wave * 16;
  int row = m0 + l;
  bool rok = row < DEC_STEPS;
  bool cok = l < NCLS;

  v8f acc = {};
  #pragma unroll
  for (int kk = 0; kk < HID; kk += 4) {
    int k0 = kk + half * 2;                    // lanes 0-15: K={kk,kk+1}; 16-31: {kk+2,kk+3}
    v2f A, B;
    A.x = rok ? Hd[row * HID + k0 + 0] : 0.f;
    A.y = rok ? Hd[row * HID + k0 + 1] : 0.f;
    B.x = cok ? linW[l * HID + k0 + 0] : 0.f;
    B.y = cok ? linW[l * HID + k0 + 1] : 0.f;
    acc = __builtin_amdgcn_wmma_f32_16x16x4_f32(false, A, false, B,
                                                (short)0, acc, false, false);
  }
  if (cok) {
    float bv = linB[l];
    #pragma unroll
    for (int r = 0; r < 8; ++r) {
      int m = m0 + r + half * 8;
      out[m * NCLS + l] = (m < DEC_STEPS) ? (acc[r] + bv) : 0.f;  // row 4095 zeroed
    }
  }
}

// ---------------------------------------------------------------------------
extern "C" void kernel_launch(void* const* d_in, const int* in_sizes, int n_in,
                              void* d_out, int out_size, void* d_ws, size_t ws_size,
                              hipStream_t stream)
{
  const float* x      = (const float*)d_in[0];
  const int*   y      = (const int*)  d_in[1];
  const float* encWih = (const float*)d_in[2];
  const float* encWhh = (const float*)d_in[3];
  const float* encBih = (const float*)d_in[4];
  const float* encBhh = (const float*)d_in[5];
  // d_in[6] = dec_Wih, [7] = dec_Whh, [8] = dec_bih, [9] = dec_bhh
  const float* decWih = (const float*)d_in[6];
  const float* decWhh = (const float*)d_in[7];
  const float* decBih = (const float*)d_in[8];
  const float* decBhh = (const float*)d_in[9];
  const float* linW   = (const float*)d_in[10];
  const float* linB   = (const float*)d_in[11];
  // d_in[12] teacher_forcing_ratio == 1 (teacher forcing always), d_in[13] n_classes == 6

  float* out = (float*)d_out;

  char* ws = (char*)d_ws;
  float* gxEnc = (float*)(ws);
  float* gxDec = (float*)(ws + (size_t)T_ENC * G4 * 4);
  float* Hdec  = (float*)(ws + (size_t)T_ENC * G4 * 4 + (size_t)T_DEC * G4 * 4);

  // 1) parallel input-contribution GEMMs (WMMA f32)
  k_gx_enc<<<800, 256, 0, stream>>>(x, encWih, encBih, gxEnc);
  int ndec = DEC_STEPS * G4;
  k_gx_dec<<<(ndec + 255) / 256, 256, 0, stream>>>(y, decWih, decBih, gxDec);
  // 2) latency-critical recurrence: one persistent WGP-resident workgroup
  k_lstm_seq<<<1, 512, 0, stream>>>(gxEnc, gxDec, encWhh, encBhh,
                                    decWhh, decBhh, Hdec);
  // 3) off-critical-path logits GEMM (WMMA f32) + zero of last output row
  k_logits<<<32, 256, 0, stream>>>(Hdec, linW, linB, out);
}